// HBP_68367289417839
// MI455X (gfx1250) — compile-verified
//
#include <hip/hip_runtime.h>
#include <math.h>

// ---- problem constants (from reference) ----
#define NN      100000
#define EE      500000
#define DD      128
#define TT      3
#define RR      4
#define HH      8
#define DKK     16
#define MAXLEN  240
#define LAYERS  2

#define GEMM_BLOCKS 384   // blocks per type; each grid-strides over its type's tiles

typedef float v2f __attribute__((ext_vector_type(2)));
typedef float v8f __attribute__((ext_vector_type(8)));

// ---------- helpers ----------
__device__ __forceinline__ float gelu_exact(float x) {
    return 0.5f * x * (1.0f + erff(x * 0.70710678118654752f));
}
// order-preserving float -> uint map (for atomicMax on signed floats)
__device__ __forceinline__ unsigned ford(float f) {
    unsigned u = __float_as_uint(f);
    return (u & 0x80000000u) ? ~u : (u | 0x80000000u);
}
__device__ __forceinline__ float iford(unsigned u) {
    return (u & 0x80000000u) ? __uint_as_float(u & 0x7fffffffu)
                             : __uint_as_float(~u);
}

// ---------- type grouping (compact rows by node type, 16-aligned) ----------
__global__ void k_count_types(const int* __restrict__ nt, int* __restrict__ counts) {
    int i = blockIdx.x * blockDim.x + threadIdx.x;
    if (i < NN) atomicAdd(&counts[nt[i]], 1);
}
__global__ void k_make_offsets(const int* __restrict__ counts,
                               int* __restrict__ offs, int* __restrict__ cursor) {
    if (threadIdx.x == 0 && blockIdx.x == 0) {
        int o = 0;
        for (int t = 0; t < TT; ++t) {
            offs[t]   = o;
            cursor[t] = o;
            o += (counts[t] + 15) & ~15;
        }
        offs[TT] = o;   // padded total rows
    }
}
__global__ void k_fill_order(const int* __restrict__ nt, int* __restrict__ cursor,
                             int* __restrict__ order) {
    int i = blockIdx.x * blockDim.x + threadIdx.x;
    if (i < NN) {
        int p = atomicAdd(&cursor[nt[i]], 1);
        order[p] = i;
    }
}

// ---------- RTE precompute ----------
// tab[pos][2D], replicating overflow-to-inf behaviour (div -> 0 for large exp)
__global__ void k_rte_tab(float* __restrict__ tab) {
    int i = blockIdx.x * blockDim.x + threadIdx.x;
    if (i >= MAXLEN * DD) return;
    int pos = i / DD, j = i % DD;
    float ar  = 2.0f * (float)j;
    float dn  = powf(10000.0f, ar) / (float)DD / 2.0f;  // inf for large ar
    float div = 1.0f / dn;                              // -> 0 on overflow
    float ang = (float)pos * div;
    const float s = 0.088388347648318447f;              // 1/sqrt(128)
    tab[pos * 2 * DD + 2 * j]     = sinf(ang) * s;
    tab[pos * 2 * DD + 2 * j + 1] = cosf(ang) * s;
}
// proj[pos][D] = tab[pos] @ rte_w + rte_b
__global__ void k_rte_proj(const float* __restrict__ tab, const float* __restrict__ rw,
                           const float* __restrict__ rb, float* __restrict__ proj) {
    int i = blockIdx.x * blockDim.x + threadIdx.x;
    if (i >= MAXLEN * DD) return;
    int p = i / DD, d = i % DD;
    float acc = rb[d];
    for (int k = 0; k < 2 * DD; ++k)
        acc = fmaf(tab[p * 2 * DD + k], rw[k * DD + d], acc);
    proj[i] = acc;
}
// ktab[t][pos][D] = proj[pos] @ W[t]   (no bias: linearity, avoids double bias)
__global__ void k_rte_kv(const float* __restrict__ proj, const float* __restrict__ Wl,
                         float* __restrict__ ktab) {
    int i = blockIdx.x * blockDim.x + threadIdx.x;
    if (i >= TT * MAXLEN * DD) return;
    int t = i / (MAXLEN * DD);
    int rem = i % (MAXLEN * DD);
    int p = rem / DD, d = rem % DD;
    const float* Wt = Wl + t * DD * DD;
    float acc = 0.0f;
    for (int k = 0; k < DD; ++k)
        acc = fmaf(proj[p * DD + k], Wt[k * DD + d], acc);
    ktab[i] = acc;
}

// ---------- elementwise exact-gelu (applied to aggr before transform GEMM) ----------
__global__ void k_gelu_inplace(float* __restrict__ a) {
    int i = blockIdx.x * blockDim.x + threadIdx.x;
    if (i < NN * DD) a[i] = gelu_exact(a[i]);
}

// ---------- typed GEMM via V_WMMA_F32_16X16X4_F32, B register-resident ----------
// Grid: (GEMM_BLOCKS, TT). Each block serves one node type: it preloads its
// 16-column slice of W[type] into 64 VGPRs of B fragments once, then
// grid-strides over that type's 16-row tiles (16-aligned compacted order).
// out[row] = epi( X[row] @ W[type(row)] + b[type(row)] )
// epi: 0 = bias, 1 = tanh(bias), 2 = sigmoid(skip)-blend with Xprev
__global__ __launch_bounds__(256)
void k_typed_gemm(const float* __restrict__ X, const int* __restrict__ order,
                  const int* __restrict__ offs, const float* __restrict__ W,
                  const float* __restrict__ Bias, const float* __restrict__ skipv,
                  const float* __restrict__ Xprev, float* __restrict__ Out,
                  int epi) {
    __shared__ __align__(16) float sA[16 * 132];   // padded stride: conflict-free
    int ty = blockIdx.y;

    int t       = threadIdx.x;
    int wave    = t >> 5;            // 8 waves -> 8 column tiles of 16
    int lane    = t & 31;
    int colBase = wave * 16;
    int mrow    = lane & 15;
    int khalf   = (lane >> 4) * 2;   // lanes 0-15: K 0,1 ; lanes 16-31: K 2,3
    int col     = colBase + mrow;

    // B fragments for this (type, column tile): resident across all row tiles
    const float* Wt = W + ty * DD * DD;
    v2f bfrag[32];
    #pragma unroll
    for (int s = 0; s < 32; ++s) {
        int k = s * 4;
        bfrag[s].x = Wt[(k + khalf)     * DD + col];
        bfrag[s].y = Wt[(k + khalf + 1) * DD + col];
    }
    float bias  = Bias[ty * DD + col];
    float alpha = 0.0f;
    if (epi == 2) alpha = 1.0f / (1.0f + expf(-skipv[ty]));

    int tile0   = offs[ty]     >> 4;
    int tileEnd = offs[ty + 1] >> 4;

    int stageR = t >> 4;             // staging: row 0..15
    int stageC = (t & 15) * 8;       // staging: 8 floats (two float4)

    for (int tile = tile0 + blockIdx.x; tile < tileEnd; tile += gridDim.x) {
        int base = tile * 16;
        __syncthreads();             // previous iteration's reads of sA complete
        {   // stage A tile (16 rows x 128 K): 2x b128 per thread
            int row = order[base + stageR];
            float4 p0 = make_float4(0.f, 0.f, 0.f, 0.f), p1 = p0;
            if (row >= 0) {
                const float4* Xv = (const float4*)(X + (size_t)row * DD + stageC);
                p0 = Xv[0];
                p1 = Xv[1];
            }
            *(float4*)&sA[stageR * 132 + stageC]     = p0;
            *(float4*)&sA[stageR * 132 + stageC + 4] = p1;
        }
        __syncthreads();

        v8f acc = {};
        #pragma unroll
        for (int s = 0; s < 32; ++s) {
            int k = s * 4;
            v2f a;
            a.x = sA[mrow * 132 + k + khalf];
            a.y = sA[mrow * 132 + k + khalf + 1];
            acc = __builtin_amdgcn_wmma_f32_16x16x4_f32(
                      false, a, false, bfrag[s], (short)0, acc, false, false);
        }

        #pragma unroll
        for (int i = 0; i < 8; ++i) {
            int m = i + (lane >> 4) * 8;       // C layout: VGPR i -> M=i / M=i+8
            int orow = order[base + m];
            if (orow < 0) continue;
            float v = acc[i] + bias;
            if (epi == 1)      v = tanhf(v);
            else if (epi == 2) v = alpha * v + (1.0f - alpha) * Xprev[orow * DD + col];
            Out[orow * DD + col] = v;
        }
    }
}

// ---------- edge pass 1: scores + segment max ----------
__global__ __launch_bounds__(256)
void k_edge_score(const float* __restrict__ Qn, const float* __restrict__ Kn,
                  const int* __restrict__ src, const int* __restrict__ tgt,
                  const int* __restrict__ etype, const int* __restrict__ etime,
                  const int* __restrict__ ntype, const float* __restrict__ ktab,
                  const float* __restrict__ rel_att, const float* __restrict__ rel_pri,
                  float* __restrict__ score, unsigned* __restrict__ segmax) {
    int e = (blockIdx.x * blockDim.x + threadIdx.x) >> 5;
    int lane = threadIdx.x & 31;
    if (e >= EE) return;
    int s = src[e], g = tgt[e], r = etype[e], tm = etime[e];
    int st = ntype[s], tt = ntype[g];
    int h = lane >> 2;                       // head: 8 heads x 4 lanes

    int d0 = lane * 4;
    float4 kq = *(const float4*)(Kn + (size_t)s * DD + d0);
    float4 kt = *(const float4*)(ktab + ((size_t)(st * MAXLEN + tm)) * DD + d0);
    float4 qq = *(const float4*)(Qn + (size_t)g * DD + d0);
    float kv[4] = {kq.x + kt.x, kq.y + kt.y, kq.z + kt.z, kq.w + kt.w};
    float qv[4] = {qq.x, qq.y, qq.z, qq.w};

    const float* A = rel_att + (r * HH + h) * DKK * DKK;
    float k2[4] = {0.f, 0.f, 0.f, 0.f};
    #pragma unroll
    for (int m = 0; m < DKK; ++m) {
        float km = __shfl(kv[m & 3], h * 4 + (m >> 2), 32);
        #pragma unroll
        for (int j = 0; j < 4; ++j) {
            int l = (lane & 3) * 4 + j;
            k2[j] = fmaf(km, A[m * DKK + l], k2[j]);
        }
    }
    float part = 0.0f;
    #pragma unroll
    for (int j = 0; j < 4; ++j) part = fmaf(qv[j], k2[j], part);
    part += __shfl_xor(part, 1, 32);
    part += __shfl_xor(part, 2, 32);
    if ((lane & 3) == 0) {
        float pri = rel_pri[((tt * RR + r) * TT + st) * HH + h];
        float sc  = part * pri * 0.25f;      // 1/sqrt(DK)
        score[e * HH + h] = sc;
        atomicMax(&segmax[g * HH + h], ford(sc));
    }
}

// ---------- edge pass 2: exp + denominator ----------
__global__ void k_edge_exp(const int* __restrict__ tgt, float* __restrict__ score,
                           const unsigned* __restrict__ segmax, float* __restrict__ den) {
    int i = blockIdx.x * blockDim.x + threadIdx.x;
    if (i >= EE * HH) return;
    int e = i / HH, h = i % HH;
    int g = tgt[e];
    float ex = expf(score[i] - iford(segmax[g * HH + h]));
    score[i] = ex;
    atomicAdd(&den[g * HH + h], ex);
}

// ---------- edge pass 3: messages + scatter ----------
__global__ __launch_bounds__(256)
void k_edge_msg(const float* __restrict__ Vn, const int* __restrict__ src,
                const int* __restrict__ tgt, const int* __restrict__ etype,
                const int* __restrict__ etime, const int* __restrict__ ntype,
                const float* __restrict__ vtab, const float* __restrict__ rel_msg,
                const float* __restrict__ score, const float* __restrict__ den,
                float* __restrict__ aggr) {
    int e = (blockIdx.x * blockDim.x + threadIdx.x) >> 5;
    int lane = threadIdx.x & 31;
    if (e >= EE) return;
    int s = src[e], g = tgt[e], r = etype[e], tm = etime[e];
    int st = ntype[s];
    int h = lane >> 2;

    int d0 = lane * 4;
    float4 va = *(const float4*)(Vn + (size_t)s * DD + d0);
    float4 vt = *(const float4*)(vtab + ((size_t)(st * MAXLEN + tm)) * DD + d0);
    float vv[4] = {va.x + vt.x, va.y + vt.y, va.z + vt.z, va.w + vt.w};

    const float* M = rel_msg + (r * HH + h) * DKK * DKK;
    float v2[4] = {0.f, 0.f, 0.f, 0.f};
    #pragma unroll
    for (int m = 0; m < DKK; ++m) {
        float vm = __shfl(vv[m & 3], h * 4 + (m >> 2), 32);
        #pragma unroll
        for (int j = 0; j < 4; ++j) {
            int l = (lane & 3) * 4 + j;
            v2[j] = fmaf(vm, M[m * DKK + l], v2[j]);
        }
    }
    float att = score[e * HH + h] / den[g * HH + h];
    #pragma unroll
    for (int j = 0; j < 4; ++j)
        atomicAdd(&aggr[g * DD + d0 + j], v2[j] * att);
}

__global__ void k_init_segmax(unsigned* __restrict__ segmax) {
    int i = blockIdx.x * blockDim.x + threadIdx.x;
    if (i < NN * HH) segmax[i] = 0x007FFFFFu;   // ford(-inf)
}

// ---------- host side ----------
extern "C" void kernel_launch(void* const* d_in, const int* in_sizes, int n_in,
                              void* d_out, int out_size, void* d_ws, size_t ws_size,
                              hipStream_t stream) {
    (void)in_sizes; (void)n_in; (void)out_size; (void)ws_size;
    const float* node_feature = (const float*)d_in[0];
    const int*   node_type    = (const int*)d_in[1];
    const int*   edge_index   = (const int*)d_in[2];
    const int*   src = edge_index;
    const int*   tgt = edge_index + EE;
    const int*   edge_type = (const int*)d_in[3];
    const int*   edge_time = (const int*)d_in[4];
    const float* adapt_w = (const float*)d_in[5];
    const float* adapt_b = (const float*)d_in[6];
    const float* Wk = (const float*)d_in[7];
    const float* bk = (const float*)d_in[8];
    const float* Wq = (const float*)d_in[9];
    const float* bq = (const float*)d_in[10];
    const float* Wv = (const float*)d_in[11];
    const float* bv = (const float*)d_in[12];
    const float* Wa = (const float*)d_in[13];
    const float* ba = (const float*)d_in[14];
    const float* rel_pri = (const float*)d_in[15];
    const float* rel_att = (const float*)d_in[16];
    const float* rel_msg = (const float*)d_in[17];
    const float* skip    = (const float*)d_in[18];
    const float* rte_w   = (const float*)d_in[19];
    const float* rte_b   = (const float*)d_in[20];
    float* out = (float*)d_out;

    // workspace carve-out (256B aligned)
    char* w = (char*)d_ws;
    auto carve = [&](size_t bytes) -> void* {
        void* p = (void*)w;
        w += (bytes + 255) & ~(size_t)255;
        return p;
    };
    float*    x0     = (float*)carve((size_t)NN * DD * 4);
    float*    x1     = (float*)carve((size_t)NN * DD * 4);
    float*    Qn     = (float*)carve((size_t)NN * DD * 4);   // aliased as aggr after use
    float*    Kn     = (float*)carve((size_t)NN * DD * 4);
    float*    Vn     = (float*)carve((size_t)NN * DD * 4);
    float*    score  = (float*)carve((size_t)EE * HH * 4);
    unsigned* segmax = (unsigned*)carve((size_t)NN * HH * 4);
    float*    den    = (float*)carve((size_t)NN * HH * 4);
    int*      order  = (int*)carve((size_t)(NN + 16 * TT) * 4);
    int*      counts = (int*)carve(TT * 4);
    int*      offs   = (int*)carve((TT + 1) * 4);
    int*      cursor = (int*)carve(TT * 4);
    float*    tab    = (float*)carve((size_t)MAXLEN * 2 * DD * 4);
    float*    proj   = (float*)carve((size_t)MAXLEN * DD * 4);
    float*    ktab   = (float*)carve((size_t)TT * MAXLEN * DD * 4);
    float*    vtab   = (float*)carve((size_t)TT * MAXLEN * DD * 4);
    float*    aggr   = Qn;   // Qn last read in edge pass 1; reuse as aggregation buffer

    const int  nodeBlocks = (NN + 255) / 256;
    const int  waveGrid   = (EE * 32 + 255) / 256;
    const dim3 gemmGrid(GEMM_BLOCKS, TT);

    // ---- build per-type compacted row order ----
    hipMemsetAsync(counts, 0, TT * 4, stream);
    k_count_types<<<nodeBlocks, 256, 0, stream>>>(node_type, counts);
    k_make_offsets<<<1, 32, 0, stream>>>(counts, offs, cursor);
    hipMemsetAsync(order, 0xFF, (size_t)(NN + 16 * TT) * 4, stream);  // -1 padding
    k_fill_order<<<nodeBlocks, 256, 0, stream>>>(node_type, cursor, order);

    // ---- RTE sinusoid table ----
    k_rte_tab<<<(MAXLEN * DD + 255) / 256, 256, 0, stream>>>(tab);

    // ---- input adaptation: x0 = tanh(feat @ adapt_w[t] + adapt_b[t]) ----
    k_typed_gemm<<<gemmGrid, 256, 0, stream>>>(node_feature, order, offs,
        adapt_w, adapt_b, nullptr, nullptr, x0, /*epi*/1);

    const float* xin = x0;
    for (int l = 0; l < LAYERS; ++l) {
        const float* Wk_l = Wk + (size_t)l * TT * DD * DD;
        const float* Wq_l = Wq + (size_t)l * TT * DD * DD;
        const float* Wv_l = Wv + (size_t)l * TT * DD * DD;
        const float* Wa_l = Wa + (size_t)l * TT * DD * DD;

        // RTE projection folded into per-(type,time) K/V tables
        k_rte_proj<<<(MAXLEN * DD + 255) / 256, 256, 0, stream>>>(
            tab, rte_w + (size_t)l * 2 * DD * DD, rte_b + (size_t)l * DD, proj);
        k_rte_kv<<<(TT * MAXLEN * DD + 255) / 256, 256, 0, stream>>>(proj, Wk_l, ktab);
        k_rte_kv<<<(TT * MAXLEN * DD + 255) / 256, 256, 0, stream>>>(proj, Wv_l, vtab);

        // Q/K/V node projections (WMMA)
        k_typed_gemm<<<gemmGrid, 256, 0, stream>>>(xin, order, offs,
            Wq_l, bq + (size_t)l * TT * DD, nullptr, nullptr, Qn, 0);
        k_typed_gemm<<<gemmGrid, 256, 0, stream>>>(xin, order, offs,
            Wk_l, bk + (size_t)l * TT * DD, nullptr, nullptr, Kn, 0);
        k_typed_gemm<<<gemmGrid, 256, 0, stream>>>(xin, order, offs,
            Wv_l, bv + (size_t)l * TT * DD, nullptr, nullptr, Vn, 0);

        // attention: score -> segment max -> exp/denominator -> weighted scatter
        k_init_segmax<<<(NN * HH + 255) / 256, 256, 0, stream>>>(segmax);
        hipMemsetAsync(den, 0, (size_t)NN * HH * 4, stream);
        k_edge_score<<<waveGrid, 256, 0, stream>>>(Qn, Kn, src, tgt, edge_type,
            edge_time, node_type, ktab,
            rel_att + (size_t)l * RR * HH * DKK * DKK,
            rel_pri + (size_t)l * TT * RR * TT * HH, score, segmax);
        hipMemsetAsync(aggr, 0, (size_t)NN * DD * 4, stream);   // Qn done; reuse
        k_edge_exp<<<(EE * HH + 255) / 256, 256, 0, stream>>>(tgt, score, segmax, den);
        k_edge_msg<<<waveGrid, 256, 0, stream>>>(Vn, src, tgt, edge_type, edge_time,
            node_type, vtab, rel_msg + (size_t)l * RR * HH * DKK * DKK,
            score, den, aggr);

        // trans = gelu(aggr) @ Wa[t] + ba[t]; x = a*trans + (1-a)*x  (WMMA, fused)
        k_gelu_inplace<<<(NN * DD + 255) / 256, 256, 0, stream>>>(aggr);
        float* xout = (l == LAYERS - 1) ? out : x1;
        k_typed_gemm<<<gemmGrid, 256, 0, stream>>>(aggr, order, offs,
            Wa_l, ba + (size_t)l * TT * DD, skip + (size_t)l * TT,
            xin, xout, /*epi*/2);
        xin = xout;
    }
}